// BeliveMapsNMS_77524159693659
// MI455X (gfx1250) — compile-verified
//
#include <hip/hip_runtime.h>
#include <stdint.h>

// Problem constants (reference: belive_map (8,2,1024,1024) fp32)
#define B_      8
#define C_      2
#define H_      1024
#define W_      1024
#define MAXP    65536
#define EX      6          // 2*MIN_DISTANCE border exclusion
#define THRESH  0.5f

#define STRIP   16         // output rows per block
#define CHUNK   256        // columns per inner chunk
#define TILE_H  (STRIP + 6)   // 22 staged rows (3-row halo each side)
#define HALO_L  4             // left halo widened to 4 so B128 loads are 16B-aligned
#define TILE_W  264           // 4 + 256 + 3 halo, padded to 264 (= 66 x 4 floats)
#define NCH16   (TILE_W / 4)  // 66 16-byte chunks per staged row

// ---------------------------------------------------------------------------
// CDNA5 async global->LDS copy (ASYNCcnt-tracked; CDNA5 ISA)
// ---------------------------------------------------------------------------
__device__ __forceinline__ void async_ld_b128(uint32_t lds_off, const float* gaddr) {
  // VDST = per-lane LDS byte offset, VADDR = 64-bit global address (GV mode)
  asm volatile("global_load_async_to_lds_b128 %0, %1, off"
               :: "v"(lds_off), "v"(gaddr) : "memory");
}
__device__ __forceinline__ void wait_async0() {
  asm volatile("s_wait_asynccnt 0" ::: "memory");
}

// ---------------------------------------------------------------------------
// Pass 1: separable 7x7 max -> peak bitmask (ballot-packed) + per-strip counts
// grid: 1024 blocks (b*128 + c*64 + strip), 256 threads (8 wave32)
// ---------------------------------------------------------------------------
__global__ void nms_mask_kernel(const float* __restrict__ bel,
                                uint32_t* __restrict__ maskbits,
                                uint32_t* __restrict__ counts) {
  __shared__ __align__(16) float tile[TILE_H][TILE_W];   // staged input rows
  __shared__ float vtile[STRIP][TILE_W];                 // vertical 7-max
  __shared__ uint32_t sred[256];

  const int blk   = blockIdx.x;
  const int b     = blk >> 7;
  const int c     = (blk >> 6) & 1;
  const int strip = blk & 63;
  const int h0    = strip * STRIP;
  const int t     = threadIdx.x;
  const int wave  = t >> 5;
  const int lane  = t & 31;
  const float* img = bel + ((size_t)(b * C_ + c) << 20);   // 1024x1024 plane

  uint32_t cnt = 0;

  for (int chunk = 0; chunk < W_ / CHUNK; ++chunk) {
    const int wbase = chunk * CHUNK;
    __syncthreads();   // prior chunk fully consumed before restaging LDS

    // Stage 22 x 264 halo tile via async B128 global->LDS. Global base cols
    // are clamped; mis-staged edge slots only feed border-excluded pixels.
    for (int idx = t; idx < TILE_H * NCH16; idx += 256) {
      const int rr = idx / NCH16;
      const int j  = idx - rr * NCH16;
      int gh = h0 - 3 + rr;           gh = gh < 0 ? 0 : (gh > H_ - 1 ? H_ - 1 : gh);
      int g0 = wbase - HALO_L + j*4;  g0 = g0 < 0 ? 0 : (g0 > W_ - 4 ? W_ - 4 : g0);
      const uint32_t loff = (uint32_t)(uintptr_t)(&tile[rr][j * 4]);
      async_ld_b128(loff, img + ((size_t)gh << 10) + g0);
    }
    // prefetch next chunk's rows into cache while asyncs drain
    if (chunk + 1 < W_ / CHUNK && t < TILE_H) {
      int gh = h0 - 3 + t; gh = gh < 0 ? 0 : (gh > H_ - 1 ? H_ - 1 : gh);
      __builtin_prefetch(img + ((size_t)gh << 10) + (wbase + CHUNK), 0, 1);
    }
    wait_async0();      // each wave waits for its own ASYNCcnt
    __syncthreads();    // all LDS writes visible block-wide

    // Vertical 7-max for all 16 output rows x 264 cols (one barrier total)
    for (int idx = t; idx < STRIP * TILE_W; idx += 256) {
      const int r  = idx / TILE_W;
      const int cc = idx - r * TILE_W;
      float m = tile[r][cc];
      #pragma unroll
      for (int k = 1; k < 7; ++k) m = fmaxf(m, tile[r + k][cc]);
      vtile[r][cc] = m;
    }
    __syncthreads();

    // Horizontal 7-max + mask: barrier-free (vtile is read-only now)
    const int cc = t + HALO_L;            // center col w = wbase + t
    const int w  = wbase + t;
    for (int r = 0; r < STRIP; ++r) {
      const int h = h0 + r;
      float hm = vtile[r][cc - 3];
      #pragma unroll
      for (int k = -2; k <= 3; ++k) hm = fmaxf(hm, vtile[r][cc + k]);
      const float val = tile[r + 3][cc];
      const bool pred = (hm == val) && (val > THRESH) &&
                        (h >= EX) && (h <= H_ - 1 - EX) &&
                        (w >= EX) && (w <= W_ - 1 - EX);

      // one ballot word per wave == 32 consecutive columns (bit i <-> w+i)
      const uint32_t bm = __builtin_amdgcn_ballot_w32(pred);
      if (lane == 0) {
        const uint32_t widx = ((uint32_t)(c * H_ + h) << 5) +
                              (uint32_t)((wbase >> 5) + wave);
        maskbits[((uint32_t)b << 16) + widx] = bm;
      }
      cnt += pred ? 1u : 0u;
    }
  }

  // block reduce peak count
  sred[t] = cnt;
  __syncthreads();
  for (int s = 128; s > 0; s >>= 1) {
    if (t < s) sred[t] += sred[t + s];
    __syncthreads();
  }
  if (t == 0) counts[blk] = sred[0];
}

// ---------------------------------------------------------------------------
// Pass 2: per-batch exclusive scan of the 128 strip counts (flat order)
// grid: 8 blocks x 128 threads
// ---------------------------------------------------------------------------
__global__ void nms_scan_kernel(const uint32_t* __restrict__ counts,
                                uint32_t* __restrict__ offsets) {
  __shared__ uint32_t s[128];
  const int b = blockIdx.x, t = threadIdx.x;
  const uint32_t v = counts[b * 128 + t];
  s[t] = v;
  __syncthreads();
  for (int d = 1; d < 128; d <<= 1) {
    const uint32_t x = (t >= d) ? s[t - d] : 0u;
    __syncthreads();
    s[t] += x;
    __syncthreads();
  }
  offsets[b * 128 + t] = s[t] - v;   // exclusive prefix within the batch
}

// ---------------------------------------------------------------------------
// Pass 3: ordered compaction from bitmask -> d_out
// grid: 1024 blocks x 32 threads (one wave32 per strip)
// ---------------------------------------------------------------------------
__global__ void nms_write_kernel(const float* __restrict__ bel,
                                 const uint32_t* __restrict__ maskbits,
                                 const uint32_t* __restrict__ offsets,
                                 float* __restrict__ out) {
  const int blk   = blockIdx.x;
  const int b     = blk >> 7;
  const int c     = (blk >> 6) & 1;
  const int strip = blk & 63;
  const int h0    = strip * STRIP;
  const int lane  = threadIdx.x;

  float* coords = out;                                   // (B, MAXP, 2)
  float* labels = out + (size_t)B_ * MAXP * 2;           // (B, MAXP)
  float* scores = out + (size_t)B_ * MAXP * 3;           // (B, MAXP)
  const float* img = bel + ((size_t)(b * C_ + c) << 20);

  uint32_t base = offsets[blk];
  for (int r = 0; r < STRIP; ++r) {
    const int h = h0 + r;
    const uint32_t word =
        maskbits[((uint32_t)b << 16) + ((uint32_t)(c * H_ + h) << 5) + lane];
    const uint32_t cnt = (uint32_t)__popc(word);

    // inclusive wave32 scan of per-word counts
    uint32_t pref = cnt;
    for (int d = 1; d < 32; d <<= 1) {
      const uint32_t x = __shfl_up(pref, d, 32);
      if (lane >= d) pref += x;
    }
    const uint32_t excl  = pref - cnt;
    const uint32_t total = __shfl(pref, 31, 32);

    uint32_t slot = base + excl;
    uint32_t m = word;
    while (m) {
      const int i = __ffs(m) - 1;    // LSB-first => increasing w => flat order
      m &= m - 1;
      const int w = (lane << 5) + i;
      if (slot < MAXP) {
        const size_t p = (size_t)b * MAXP + slot;
        coords[p * 2 + 0] = (float)w;        // x
        coords[p * 2 + 1] = (float)h;        // y
        labels[p] = (float)(c + 1);
        scores[p] = img[((size_t)h << 10) + w];
      }
      ++slot;
    }
    base += total;
  }
}

// ---------------------------------------------------------------------------
// Init: default slots (coords=-1, labels=0, scores=0); harness poisons d_out
// ---------------------------------------------------------------------------
__global__ void nms_init_kernel(float* __restrict__ out, int n, int coord_end) {
  const int i = blockIdx.x * blockDim.x + threadIdx.x;
  if (i < n) out[i] = (i < coord_end) ? -1.0f : 0.0f;
}

// ---------------------------------------------------------------------------
extern "C" void kernel_launch(void* const* d_in, const int* in_sizes, int n_in,
                              void* d_out, int out_size, void* d_ws, size_t ws_size,
                              hipStream_t stream) {
  (void)in_sizes; (void)n_in; (void)ws_size;
  const float* bel = (const float*)d_in[0];
  float* out = (float*)d_out;

  uint32_t* maskbits = (uint32_t*)d_ws;                       // 524288 words (2 MB)
  uint32_t* counts   = maskbits + (B_ * C_ * H_ * W_ / 32);   // 1024 words
  uint32_t* offsets  = counts + 1024;                         // 1024 words

  const int nblk_init = (out_size + 255) / 256;
  nms_init_kernel<<<nblk_init, 256, 0, stream>>>(out, out_size, B_ * MAXP * 2);
  nms_mask_kernel<<<B_ * C_ * (H_ / STRIP), 256, 0, stream>>>(bel, maskbits, counts);
  nms_scan_kernel<<<B_, 128, 0, stream>>>(counts, offsets);
  nms_write_kernel<<<B_ * C_ * (H_ / STRIP), 32, 0, stream>>>(bel, maskbits, offsets, out);
}